// ClustGeoNodeEncoder_63608465654085
// MI455X (gfx1250) — compile-verified
//
#include <hip/hip_runtime.h>
#include <math.h>

// MI455X / gfx1250, wave32. One wave per cluster; WMMA f32 16x16x4 computes
// the 3x3 Gram (scatter) matrix on the matrix pipe. Branchless LDS addressing
// with a zero-pad region + 1-deep software pipeline in the WMMA loop.

typedef float v2f __attribute__((ext_vector_type(2)));
typedef float v8f __attribute__((ext_vector_type(8)));

#define L_MAX 256
#define CL_PER_BLOCK 8          // 8 waves (wave32) per 256-thread block
#define XC_STRIDE   (L_MAX * 3) // 768 floats per wave slice
#define XC_TOTAL    (CL_PER_BLOCK * XC_STRIDE)   // 6144
#define ZPAD_OFF    XC_TOTAL                      // 48 zero floats
#define ZPAD_N      48
#define SOUT_OFF    (ZPAD_OFF + ZPAD_N)           // 6192
#define NOUT        19
#define SOUT_N      (CL_PER_BLOCK * NOUT)         // 152
#define SMEM_N      (SOUT_OFF + SOUT_N)           // 6344 floats (~25.4 KB)

__device__ __forceinline__ float wave_sum(float v) {
  #pragma unroll
  for (int m = 16; m > 0; m >>= 1) v += __shfl_xor(v, m, 32);
  return v; // butterfly: every lane holds the total
}

// Eigenvector of symmetric A for eigenvalue w via largest row-cross of (A - wI).
__device__ __forceinline__ void eig_vec(float a00, float a01, float a02,
                                        float a11, float a12, float a22,
                                        float w, float& vx, float& vy, float& vz) {
  float r0x = a00 - w, r0y = a01,     r0z = a02;
  float r1x = a01,     r1y = a11 - w, r1z = a12;
  float r2x = a02,     r2y = a12,     r2z = a22 - w;
  float c0x = r0y*r1z - r0z*r1y, c0y = r0z*r1x - r0x*r1z, c0z = r0x*r1y - r0y*r1x;
  float c1x = r0y*r2z - r0z*r2y, c1y = r0z*r2x - r0x*r2z, c1z = r0x*r2y - r0y*r2x;
  float c2x = r1y*r2z - r1z*r2y, c2y = r1z*r2x - r1x*r2z, c2z = r1x*r2y - r1y*r2x;
  float n0 = c0x*c0x + c0y*c0y + c0z*c0z;
  float n1 = c1x*c1x + c1y*c1y + c1z*c1z;
  float n2 = c2x*c2x + c2y*c2y + c2z*c2z;
  float bx = c0x, by = c0y, bz = c0z, bn = n0;
  if (n1 > bn) { bx = c1x; by = c1y; bz = c1z; bn = n1; }
  if (n2 > bn) { bx = c2x; by = c2y; bz = c2z; bn = n2; }
  if (bn < 1e-30f) { vx = 0.f; vy = 0.f; vz = 1.f; return; }
  float inv = 1.0f / sqrtf(bn);
  vx = bx * inv; vy = by * inv; vz = bz * inv;
}

__global__ __launch_bounds__(256)
void ClustGeoNodeEncoder_kernel(const float* __restrict__ data,
                                const int*   __restrict__ clust_idx,
                                const int*   __restrict__ clust_len,
                                float*       __restrict__ out,
                                int C) {
  __shared__ float smem[SMEM_N];
  const int wave = threadIdx.x >> 5;
  const int lane = threadIdx.x & 31;
  const int c_raw = blockIdx.x * CL_PER_BLOCK + wave;
  const bool active = c_raw < C;
  const int c = active ? c_raw : (C - 1);   // clamp: uniform control flow, no early return

  // zero the WMMA pad region (read by lanes with m >= 3)
  if (threadIdx.x < ZPAD_N) smem[ZPAD_OFF + threadIdx.x] = 0.0f;

  const int   len = clust_len[c];
  const float n   = (float)len;
  const int* idxp = clust_idx + (size_t)c * L_MAX;
  float* wlds = &smem[wave * XC_STRIDE];

  // ---------- pass 1: gather + first-moment accumulation ----------
  float px[8], py[8], pz[8];
  float sx = 0.f, sy = 0.f, sz = 0.f, sv = 0.f, svv = 0.f;
  float h0 = 0.f, h1 = 0.f, h2 = 0.f, h3 = 0.f, h4 = 0.f;
  #pragma unroll
  for (int it = 0; it < 8; ++it) {
    int  l     = lane + 32 * it;
    bool valid = l < len;
    int  idx   = idxp[l];
    const float* row = data + (size_t)idx * 6;
    float  x  = row[0], y = row[1], z = row[2];
    float2 vs = *reinterpret_cast<const float2*>(row + 4); // 8B-aligned (idx*24+16)
    float  v  = vs.x, s = vs.y;
    if (!valid) { x = 0.f; y = 0.f; z = 0.f; v = 0.f; s = -1.f; }
    px[it] = x; py[it] = y; pz[it] = z;
    sx += x; sy += y; sz += z; sv += v; svv += v * v;
    int si = (int)s;
    h0 += (si == 0); h1 += (si == 1); h2 += (si == 2); h3 += (si == 3); h4 += (si == 4);
  }
  sx = wave_sum(sx);  sy = wave_sum(sy);  sz = wave_sum(sz);
  sv = wave_sum(sv);  svv = wave_sum(svv);
  h0 = wave_sum(h0);  h1 = wave_sum(h1);  h2 = wave_sum(h2);
  h3 = wave_sum(h3);  h4 = wave_sum(h4);

  const float invn = 1.0f / n;
  const float mx = sx * invn, my = sy * invn, mz = sz * invn;
  const float mean_v = sv * invn;
  float var_v = (svv - n * mean_v * mean_v) / (n - 1.0f);
  const float std_v = sqrtf(fmaxf(var_v, 0.0f));

  float best = h0, mode = 0.f;
  if (h1 > best) { best = h1; mode = 1.f; }
  if (h2 > best) { best = h2; mode = 2.f; }
  if (h3 > best) { best = h3; mode = 3.f; }
  if (h4 > best) { best = h4; mode = 4.f; }

  // ---------- pass 1b: centered + masked coords -> registers and LDS ----------
  #pragma unroll
  for (int it = 0; it < 8; ++it) {
    int  l     = lane + 32 * it;
    bool valid = l < len;
    float cx = valid ? (px[it] - mx) : 0.f;
    float cy = valid ? (py[it] - my) : 0.f;
    float cz = valid ? (pz[it] - mz) : 0.f;
    px[it] = cx; py[it] = cy; pz[it] = cz;
    wlds[l * 3 + 0] = cx;
    wlds[l * 3 + 1] = cy;
    wlds[l * 3 + 2] = cz;
  }
  __syncthreads();  // zpad + all wave slices visible

  // ---------- Gram matrix via V_WMMA_F32_16X16X4_F32 ----------
  // A(16x4): row m = coord m of 4 points; B(4x16): col n = coord n of the same
  // 4 points. With the ISA lane layouts each lane's A and B register contents
  // are identical (m = n = lane%16, K-pair (0,1)/(2,3) by half-wave), so one
  // 2-VGPR operand feeds both. Lanes m>=3 read a zero-pad region (distinct
  // per-lane words, advance stride 0) -> fully branchless, no EXEC juggling.
  const int  m  = lane & 15;
  const bool hi = lane >= 16;
  // index relative to wlds; pad lanes resolve to smem[ZPAD_OFF + lane]
  int i0 = (m < 3) ? ((hi ? 2 : 0) * 3 + m)
                   : (ZPAD_OFF - wave * XC_STRIDE + lane);
  const int step = (m < 3) ? 12 : 0;   // 4 points * 3 floats per K-group

  float ax = wlds[i0];
  float ay = wlds[i0 + 3];
  v8f acc = {0.f, 0.f, 0.f, 0.f, 0.f, 0.f, 0.f, 0.f};
  #pragma unroll 4
  for (int g = 0; g < 64; ++g) {
    v2f a = {ax, ay};
    i0 += step;
    ax = wlds[i0];        // prefetch next group (g=63 over-read stays in smem,
    ay = wlds[i0 + 3];    //  value discarded)
    acc = __builtin_amdgcn_wmma_f32_16x16x4_f32(
        /*neg_a=*/false, a, /*neg_b=*/false, a,
        /*c_mod=*/(short)0, acc, /*reuse_a=*/false, /*reuse_b=*/false);
  }
  // C/D layout: lanes 0-15, VGPR r -> D[M=r][N=lane]. Broadcast the 6 uniques.
  float a00 = __shfl(acc[0], 0, 32);
  float a01 = __shfl(acc[0], 1, 32);
  float a02 = __shfl(acc[0], 2, 32);
  float a11 = __shfl(acc[1], 1, 32);
  float a12 = __shfl(acc[1], 2, 32);
  float a22 = __shfl(acc[2], 2, 32);

  // ---------- analytic symmetric 3x3 eigensolve (all lanes redundantly) ----------
  float q  = (a00 + a11 + a22) * (1.0f / 3.0f);
  float p1 = a01 * a01 + a02 * a02 + a12 * a12;
  float b00 = a00 - q, b11 = a11 - q, b22 = a22 - q;
  float p2 = b00 * b00 + b11 * b11 + b22 * b22 + 2.0f * p1;
  float w0, w1, w2;
  if (p2 <= 1e-30f) {
    w0 = q; w1 = q; w2 = q;
  } else {
    float p = sqrtf(p2 * (1.0f / 6.0f));
    float ip = 1.0f / p;
    float c00 = b00 * ip, c01 = a01 * ip, c02 = a02 * ip;
    float c11 = b11 * ip, c12 = a12 * ip, c22 = b22 * ip;
    float detB = c00 * (c11 * c22 - c12 * c12)
               - c01 * (c01 * c22 - c12 * c02)
               + c02 * (c01 * c12 - c11 * c02);
    float r = fminf(1.0f, fmaxf(-1.0f, 0.5f * detB));
    float phi = acosf(r) * (1.0f / 3.0f);
    w2 = q + 2.0f * p * cosf(phi);
    w0 = q + 2.0f * p * cosf(phi + 2.0943951023931953f); // + 2*pi/3
    w1 = 3.0f * q - w0 - w2;
  }

  float v2x, v2y, v2z, u0x, u0y, u0z;
  eig_vec(a00, a01, a02, a11, a12, a22, w2, v2x, v2y, v2z);
  eig_vec(a00, a01, a02, a11, a12, a22, w0, u0x, u0y, u0z);
  // orthogonalize u0 against v2 for robustness
  float dp = u0x * v2x + u0y * v2y + u0z * v2z;
  u0x -= dp * v2x; u0y -= dp * v2y; u0z -= dp * v2z;
  float un = u0x * u0x + u0y * u0y + u0z * u0z;
  if (un < 1e-24f) {
    float ex = (fabsf(v2x) < 0.9f) ? 1.f : 0.f;
    float ey = 1.f - ex;
    u0x = -v2z * ey;
    u0y =  v2z * ex;
    u0z =  v2x * ey - v2y * ex;
    un  = fmaxf(u0x * u0x + u0y * u0y + u0z * u0z, 1e-30f);
  }
  float iun = 1.0f / sqrtf(un);
  u0x *= iun; u0y *= iun; u0z *= iun;
  float v1x = v2y * u0z - v2z * u0y;
  float v1y = v2z * u0x - v2x * u0z;
  float v1z = v2x * u0y - v2y * u0x;

  float iw2 = 1.0f / w2;                 // matches reference NaN behavior at w2==0
  float wd0 = w0 * iw2, wd1 = w1 * iw2, wd2 = w2 * iw2;
  float dirwt = (w2 == 0.0f) ? 0.0f : (1.0f - w1 * iw2);

  // ---------- pass 2: sign scalar sc from register-resident centered coords ----------
  float sc = 0.f;
  #pragma unroll
  for (int it = 0; it < 8; ++it) {
    float cx = px[it], cy = py[it], cz = pz[it];
    float x0 = cx * v2x + cy * v2y + cz * v2z;
    float rx = cx - x0 * v2x;
    float ry = cy - x0 * v2y;
    float rz = cz - x0 * v2z;
    float sq = rx * rx + ry * ry + rz * rz;
    sc += x0 * sqrtf(sq);               // masked points are exactly 0 -> contribute 0
  }
  sc = wave_sum(sc);

  float sgn = (sc < 0.0f) ? -dirwt : dirwt;
  float o0x = v2x * sgn, o0y = v2y * sgn, o0z = v2z * sgn;

  // B = wd0*u0 u0^T + wd1*v1 v1^T + wd2*v2 v2^T  (symmetric)
  float B00 = wd0*u0x*u0x + wd1*v1x*v1x + wd2*v2x*v2x;
  float B01 = wd0*u0x*u0y + wd1*v1x*v1y + wd2*v2x*v2y;
  float B02 = wd0*u0x*u0z + wd1*v1x*v1z + wd2*v2x*v2z;
  float B11 = wd0*u0y*u0y + wd1*v1y*v1y + wd2*v2y*v2y;
  float B12 = wd0*u0y*u0z + wd1*v1y*v1z + wd2*v2y*v2z;
  float B22 = wd0*u0z*u0z + wd1*v1z*v1z + wd2*v2z*v2z;

  // ---------- stage outputs in LDS, then block-coalesced global store ----------
  if (lane == 0) {
    float* o = &smem[SOUT_OFF + wave * NOUT];
    o[0]  = mx;  o[1]  = my;  o[2]  = mz;
    o[3]  = B00; o[4]  = B01; o[5]  = B02;
    o[6]  = B01; o[7]  = B11; o[8]  = B12;
    o[9]  = B02; o[10] = B12; o[11] = B22;
    o[12] = o0x; o[13] = o0y; o[14] = o0z;
    o[15] = n;
    o[16] = mean_v; o[17] = std_v; o[18] = mode;
  }
  __syncthreads();
  int t = threadIdx.x;
  if (t < SOUT_N) {
    long long gi = (long long)blockIdx.x * SOUT_N + t;
    if (gi < (long long)C * NOUT) out[gi] = smem[SOUT_OFF + t];
  }
}

extern "C" void kernel_launch(void* const* d_in, const int* in_sizes, int n_in,
                              void* d_out, int out_size, void* d_ws, size_t ws_size,
                              hipStream_t stream) {
  const float* data      = (const float*)d_in[0];
  const int*   clust_idx = (const int*)d_in[1];
  const int*   clust_len = (const int*)d_in[2];
  float*       out       = (float*)d_out;
  (void)n_in; (void)out_size; (void)d_ws; (void)ws_size;

  const int C = in_sizes[2];                       // one length per cluster
  const int blocks = (C + CL_PER_BLOCK - 1) / CL_PER_BLOCK;
  hipLaunchKernelGGL(ClustGeoNodeEncoder_kernel, dim3(blocks), dim3(256), 0, stream,
                     data, clust_idx, clust_len, out, C);
}